// MoNet_30279519437692
// MI455X (gfx1250) — compile-verified
//
#include <hip/hip_runtime.h>
#include <math.h>

typedef __attribute__((ext_vector_type(2))) float v2f;
typedef __attribute__((ext_vector_type(8))) float v8f;

#define NN   20000
#define NE   320000
#define DOUT 128
#define NOUT 512   // K * DOUT = 4 * 128

// ---------------------------------------------------------------------------
// GEMM: C[M,512] = A[M,KD] * B[KD,512], fp32 via V_WMMA_F32_16X16X4_F32.
// Block = 256 threads (8 waves) owns one 16-row M panel (grid = M/16 = 1250).
// A tile (16 x KD) staged once in LDS (row stride KD+2 -> conflict-free b64
// reads). Each wave computes 4 adjacent 16x16 N tiles (16x64), reusing the A
// fragment across 4 WMMAs per k-step.
//
// WMMA f32 16x16x4 operand layout (wave32):
//   A: lane L -> row M=L%16; VGPR0 = K {0|2}, VGPR1 = K {1|3}  (kin = 2*(L/16))
//   B: lane L -> col N=L%16; VGPR0 = K-row {0|2}, VGPR1 = {1|3}
//   C/D: VGPR i = rows (i, i+8); lane L = col L%16, half L/16
// ---------------------------------------------------------------------------
template <int KD>
__global__ __launch_bounds__(256) void gemm_wmma_lds(
    const float* __restrict__ A, const float* __restrict__ B,
    float* __restrict__ C) {
  __shared__ float As[16 * (KD + 2)];
  const int tm  = blockIdx.x;
  const int tid = threadIdx.x;

  // Cooperative coalesced load of the 16 x KD A tile (float4 chunks; 4 | KD
  // so chunks never straddle rows).
  const float* atile = A + (size_t)tm * 16 * KD;
  constexpr int NCHUNK = 16 * KD / 4;
#pragma unroll
  for (int c = tid; c < NCHUNK; c += 256) {
    const float4 v = *(const float4*)(atile + c * 4);
    const int rowi = (c * 4) / KD;
    const int coli = (c * 4) % KD;
    float* dp = &As[rowi * (KD + 2) + coli];
    dp[0] = v.x; dp[1] = v.y; dp[2] = v.z; dp[3] = v.w;
  }
  __syncthreads();

  const int lane = tid & 31;
  const int wave = tid >> 5;
  const int r    = lane & 15;
  const int hi   = lane >> 4;   // 0 | 1
  const int kin  = hi * 2;      // K sub-offset 0 | 2
  const int col0 = wave * 64 + r;  // wave covers cols [wave*64, wave*64+64)

  v8f acc0 = {}, acc1 = {}, acc2 = {}, acc3 = {};
  for (int k0 = 0; k0 < KD; k0 += 4) {
    const v2f a = *(const v2f*)(&As[r * (KD + 2) + k0 + kin]);  // 8B aligned
    const float* b0p = B + (size_t)(k0 + kin) * NOUT + col0;
    const float* b1p = b0p + NOUT;
    v2f b0, b1, b2, b3;
    b0.x = b0p[0];  b0.y = b1p[0];
    b1.x = b0p[16]; b1.y = b1p[16];
    b2.x = b0p[32]; b2.y = b1p[32];
    b3.x = b0p[48]; b3.y = b1p[48];
    acc0 = __builtin_amdgcn_wmma_f32_16x16x4_f32(false, a, false, b0, (short)0, acc0, false, false);
    acc1 = __builtin_amdgcn_wmma_f32_16x16x4_f32(false, a, false, b1, (short)0, acc1, false, false);
    acc2 = __builtin_amdgcn_wmma_f32_16x16x4_f32(false, a, false, b2, (short)0, acc2, false, false);
    acc3 = __builtin_amdgcn_wmma_f32_16x16x4_f32(false, a, false, b3, (short)0, acc3, false, false);
  }

  float* crow = C + (size_t)(tm * 16 + hi * 8) * NOUT + col0;
#pragma unroll
  for (int i = 0; i < 8; ++i) {
    crow[i * NOUT +  0] = acc0[i];
    crow[i * NOUT + 16] = acc1[i];
    crow[i * NOUT + 32] = acc2[i];
    crow[i * NOUT + 48] = acc3[i];
  }
}

// ---------------------------------------------------------------------------
// Per-edge GMM weights: u = tanh(pseudo @ pw + pb); w[e,k] = exp(-.5*|(u-mu_k)*is_k|^2)
// ---------------------------------------------------------------------------
__global__ void edge_weights(const float* __restrict__ pseudo,
                             const float* __restrict__ pw,
                             const float* __restrict__ pb,
                             const float* __restrict__ mu,
                             const float* __restrict__ isg,
                             float* __restrict__ wbuf, int E) {
  int e = blockIdx.x * blockDim.x + threadIdx.x;
  if (e >= E) return;
  float p0 = pseudo[2 * e + 0];
  float p1 = pseudo[2 * e + 1];
  float u0 = tanhf(p0 * pw[0] + p1 * pw[2] + pb[0]);
  float u1 = tanhf(p0 * pw[1] + p1 * pw[3] + pb[1]);
#pragma unroll
  for (int k = 0; k < 4; ++k) {
    float d0 = (u0 - mu[2 * k + 0]) * isg[2 * k + 0];
    float d1 = (u1 - mu[2 * k + 1]) * isg[2 * k + 1];
    wbuf[4 * e + k] = expf(-0.5f * (d0 * d0 + d1 * d1));
  }
}

// out[n,o] = bias[o]  (also clears the 0xAA-poisoned accumulator)
__global__ void init_bias(float* __restrict__ out,
                          const float* __restrict__ bias, int total) {
  int t = blockIdx.x * blockDim.x + threadIdx.x;
  if (t < total) out[t] = bias[t & (DOUT - 1)];
}

// ---------------------------------------------------------------------------
// Gather-mix-scatter: 32 lanes per edge, 4 outputs per lane (float4):
//   m[0:4] = sum_k w[e,k] * hp[src[e], k, 4*lane : 4*lane+4]
//   atomicAdd x4 into out[dst[e]].
// hp (41 MB) is L2-resident on MI455X (192 MB L2) -> gathers hit L2.
// ---------------------------------------------------------------------------
__global__ void edge_scatter(const float* __restrict__ hp,
                             const float* __restrict__ wbuf,
                             const int* __restrict__ src,
                             const int* __restrict__ dst,
                             float* __restrict__ out, int E) {
  int t  = blockIdx.x * blockDim.x + threadIdx.x;
  int e  = t >> 5;          // 32 lanes per edge
  int qi = t & 31;          // float4 index within the 128-wide output
  if (e >= E) return;
  int s = src[e], d = dst[e];
  const float4* h = (const float4*)(hp + (size_t)s * NOUT);
  float4 h0 = h[qi], h1 = h[32 + qi], h2 = h[64 + qi], h3 = h[96 + qi];
  float w0 = wbuf[4 * e + 0], w1 = wbuf[4 * e + 1];
  float w2 = wbuf[4 * e + 2], w3 = wbuf[4 * e + 3];
  float mx = w0 * h0.x + w1 * h1.x + w2 * h2.x + w3 * h3.x;
  float my = w0 * h0.y + w1 * h1.y + w2 * h2.y + w3 * h3.y;
  float mz = w0 * h0.z + w1 * h1.z + w2 * h2.z + w3 * h3.z;
  float mw = w0 * h0.w + w1 * h1.w + w2 * h2.w + w3 * h3.w;
  float* o = out + (size_t)d * DOUT + qi * 4;
  atomicAdd(o + 0, mx);
  atomicAdd(o + 1, my);
  atomicAdd(o + 2, mz);
  atomicAdd(o + 3, mw);
}

// ---------------------------------------------------------------------------
extern "C" void kernel_launch(void* const* d_in, const int* in_sizes, int n_in,
                              void* d_out, int out_size, void* d_ws, size_t ws_size,
                              hipStream_t stream) {
  const float* feat   = (const float*)d_in[0];
  const float* pseudo = (const float*)d_in[1];
  const int*   src    = (const int*)d_in[2];
  const int*   dst    = (const int*)d_in[3];

  // Workspace layout (fp32): hp | edge weights | hA | hB   (~67 MB total)
  float* hp = (float*)d_ws;
  float* wb = hp + (size_t)NN * NOUT;
  float* hA = wb + (size_t)NE * 4;
  float* hB = hA + (size_t)NN * DOUT;

  const float* hin = feat;
  float* houts[3] = { hA, hB, (float*)d_out };

  for (int L = 0; L < 3; ++L) {
    const float* fc   = (const float*)d_in[4 + 6 * L + 0];
    const float* mu   = (const float*)d_in[4 + 6 * L + 1];
    const float* isg  = (const float*)d_in[4 + 6 * L + 2];
    const float* bias = (const float*)d_in[4 + 6 * L + 3];
    const float* pw   = (const float*)d_in[4 + 6 * L + 4];
    const float* pb   = (const float*)d_in[4 + 6 * L + 5];
    float* hout = houts[L];

    if (L == 0)
      gemm_wmma_lds<64><<<NN / 16, 256, 0, stream>>>(hin, fc, hp);
    else
      gemm_wmma_lds<128><<<NN / 16, 256, 0, stream>>>(hin, fc, hp);

    edge_weights<<<(NE + 255) / 256, 256, 0, stream>>>(pseudo, pw, pb, mu, isg, wb, NE);
    init_bias<<<(NN * DOUT + 255) / 256, 256, 0, stream>>>(hout, bias, NN * DOUT);
    edge_scatter<<<(NE * 32) / 256, 256, 0, stream>>>(hp, wb, src, dst, hout, NE);

    hin = hout;
  }
}